// CustomComputationLayer_87368224735398
// MI455X (gfx1250) — compile-verified
//
#include <hip/hip_runtime.h>
#include <hip/hip_bf16.h>
#include <stdint.h>

// ---------------------------------------------------------------------------
// Problem constants (mirror the JAX reference)
// ---------------------------------------------------------------------------
#define T_STEPS 20
#define ROW_F   (5 * T_STEPS)   // 100 floats per row
#define ROW_Q   (ROW_F / 4)     // 25 float4 per row
#define RPB     128             // rows (threads) per block
// LDS per block: RPB * ROW_F * 4 = 51,200 B  (fits static-shared; WGP has 320KB)

// Vector-of-4-int type matching the async-LDS builtin's expected pointee
typedef int v4i __attribute__((vector_size(4 * sizeof(int))));
typedef __attribute__((address_space(1))) v4i* gv4i_p;   // global (AS1)
typedef __attribute__((address_space(3))) v4i* lv4i_p;   // LDS    (AS3)

// Compile-time tables: PBARD[t] = (1 - cumsum(pd)[t]) * 0.995^t ; PD[t] = pd[t]*0.995^t
// pd = linspace(0.001, 0.01, 21)  (double math, then cast to f32 — matches numpy)
struct Tables {
    float pbard[T_STEPS + 1];
    float pd[T_STEPS];
    constexpr Tables() : pbard{}, pd{} {
        double D = 1.0, cum = 0.0;
        for (int t = 0; t <= T_STEPS; ++t) {
            double pdt = 0.001 + (0.009 / 20.0) * (double)t;
            cum += pdt;
            pbard[t] = (float)((1.0 - cum) * D);
            if (t < T_STEPS) pd[t] = (float)(pdt * D);
            D *= 0.995;
        }
    }
};
__constant__ Tables g_tab = Tables();

// ---------------------------------------------------------------------------
// Kernel: one thread == one row. Block stages 128 rows (50 KB) into LDS with
// CDNA5 async global->LDS b128 copies, waits on ASYNCcnt, then each thread
// walks its row out of LDS (ds_load_b128 per timestep) doing the scan.
// ---------------------------------------------------------------------------
__global__ __launch_bounds__(RPB)
void econ_scan_kernel(const float* __restrict__ in, float* __restrict__ out, int nrows)
{
    __shared__ float4 s_buf[RPB * ROW_Q];   // 51,200 bytes

    const int tid      = threadIdx.x;
    const int blockRow = blockIdx.x * RPB;

    // ---- Stage RPB rows, fully coalesced: linear float4 copy --------------
    const float4* gsrc = (const float4*)(in + (size_t)blockRow * ROW_F);

#if defined(__gfx1250__) && __has_builtin(__builtin_amdgcn_global_load_async_to_lds_b128)
    {
        #pragma unroll
        for (int it = 0; it < ROW_Q; ++it) {
            const int idx = it * RPB + tid;          // float4 index within block tile
            __builtin_amdgcn_global_load_async_to_lds_b128(
                (gv4i_p)(gsrc + idx),
                (lv4i_p)(s_buf + idx),
                /*offset=*/0, /*cpol=*/0);
        }
    }
  #if __has_builtin(__builtin_amdgcn_s_wait_asynccnt)
    __builtin_amdgcn_s_wait_asynccnt(0);
  #else
    asm volatile("s_wait_asynccnt 0" ::: "memory");
  #endif
#else
    // Fallback: coalesced b128 load -> LDS store
    #pragma unroll
    for (int it = 0; it < ROW_Q; ++it) {
        const int idx = it * RPB + tid;
        s_buf[idx] = gsrc[idx];
    }
#endif
    __syncthreads();

    // ---- Per-row compute ---------------------------------------------------
    const float*  rowf = (const float*)(s_buf) + tid * ROW_F;
    const float4* rowq = (const float4*)rowf;

    const float R_RATE = 0.02f;
    float x     = 1.0f;     // x[0]
    float total = 0.0f;

    #pragma unroll
    for (int t = 0; t < T_STEPS; ++t) {
        const float4 q    = rowq[t];                 // cv, bv, iv, rhov (ds_load_b128)
        const float  Irel = rowf[4 * T_STEPS + t];

        // softmax over (cv, bv, iv), max-subtracted like jax.nn.softmax
        const float m   = fmaxf(q.x, fmaxf(q.y, q.z));
        const float ec  = __expf(q.x - m);
        const float eb  = __expf(q.y - m);
        const float ei  = __expf(q.z - m);
        const float inv = __builtin_amdgcn_rcpf(ec + eb + ei);
        float c = fminf(fmaxf(ec * inv, 1e-6f), 1.0f - 1e-6f);
        float b = fminf(fmaxf(eb * inv, 1e-6f), 1.0f - 1e-6f);

        // sigmoid(rhov)
        float rho = __builtin_amdgcn_rcpf(1.0f + __expf(-q.w));
        rho = fminf(fmaxf(rho, 1e-6f), 1.0f - 1e-6f);

        float wf = 1.0f - c - b + R_RATE + rho * Irel;
        wf = fminf(fmaxf(wf, 1e-6f), 1e6f);

        const float pb  = g_tab.pbard[t];
        const float pdt = g_tab.pd[t];

        // consumption utility: vc(c * x_t) * PBARD[t],  vc(C)=C^0.2/0.2
        float C = fminf(fmaxf(c * x, 1e-6f), 1e6f);
        total += __expf(0.2f * __logf(C)) * 5.0f * pb;

        // legacy utility: vx(x_t * pd * (1 + b/(pd+1e-6))),  vx(X)=X^0.9/9
        float L = x * pdt * (1.0f + b * __builtin_amdgcn_rcpf(pdt + 1e-6f));
        L = fminf(fmaxf(L, 1e-6f), 1e6f);
        total += __expf(0.9f * __logf(L)) * (1.0f / 9.0f);

        // scan: x_{t+1} = x_t * wf_t + 1
        x = x * wf + 1.0f;
    }

    // terminal: vx(x[T]) * PBARD[T]
    float XF = fminf(fmaxf(x, 1e-6f), 1e6f);
    total += __expf(0.9f * __logf(XF)) * (1.0f / 9.0f) * g_tab.pbard[T_STEPS];

    total *= 1.0f / (float)(T_STEPS * T_STEPS);   // UTILITY_FACTOR = 1/400

    const int grow = blockRow + tid;
    if (grow < nrows) out[grow] = total;
}

// ---------------------------------------------------------------------------
extern "C" void kernel_launch(void* const* d_in, const int* in_sizes, int n_in,
                              void* d_out, int out_size, void* d_ws, size_t ws_size,
                              hipStream_t stream)
{
    (void)n_in; (void)d_ws; (void)ws_size; (void)out_size;
    const float* in  = (const float*)d_in[0];
    float*       out = (float*)d_out;

    const int nrows = in_sizes[0] / ROW_F;          // 1048576
    const int grid  = (nrows + RPB - 1) / RPB;      // 8192 blocks

    econ_scan_kernel<<<dim3(grid), dim3(RPB), 0, stream>>>(in, out, nrows);
}